// LightGCN_17471926960600
// MI455X (gfx1250) — compile-verified
//
#include <hip/hip_runtime.h>

#define NUM_USERS 100000
#define NUM_ITEMS 100000
#define EMB       64
#define N_NODES   (NUM_USERS + NUM_ITEMS)
#define NUM_LAYERS 3

#define CHUNK 1024   // edges staged per block per step
#define TPB   256

typedef float v2f __attribute__((ext_vector_type(2)));
typedef float v8f __attribute__((ext_vector_type(8)));

// ---------------- CDNA5 async global->LDS staging ----------------
#if __has_builtin(__builtin_amdgcn_global_load_async_to_lds_b128) && \
    __has_builtin(__builtin_amdgcn_s_wait_asynccnt)
#define HAVE_ASYNC 1
// Builtin signature (from clang diagnostic): param0 = addrspace(1) ptr to
// 'int __attribute__((vector_size(16)))' (global source), param1 = LDS dest.
typedef int vi4 __attribute__((vector_size(16)));
typedef __attribute__((address_space(1))) vi4 g_vi4;
typedef __attribute__((address_space(3))) vi4 l_vi4;
#else
#define HAVE_ASYNC 0
#endif

__device__ __forceinline__ void async_cp16(void* lds_dst, const void* g_src) {
#if HAVE_ASYNC
  __builtin_amdgcn_global_load_async_to_lds_b128((g_vi4*)g_src, (l_vi4*)lds_dst,
                                                 0, 0);
#else
  *(int4*)lds_dst = *(const int4*)g_src;
#endif
}

template <int N>
__device__ __forceinline__ void async_wait() {
#if HAVE_ASYNC
  __builtin_amdgcn_s_wait_asynccnt(N);
#endif
}

// Hardware f32 atomic add (global_atomic_add_f32, no-return form when result
// is unused).  unsafeAtomicAdd is HIP's per-arch native FP atomic selector.
__device__ __forceinline__ void atom_add_f32(float* p, float v) {
#if defined(__HIP_PLATFORM_AMD__) || defined(__HIP_DEVICE_COMPILE__)
  unsafeAtomicAdd(p, v);
#else
  __hip_atomic_fetch_add(p, v, __ATOMIC_RELAXED, __HIP_MEMORY_SCOPE_AGENT);
#endif
}

// ---------------- init: out = x_cur = concat(user_emb, item_emb) ----------------
__global__ void __launch_bounds__(TPB) lgcn_init(const float* __restrict__ user_emb,
                                                 const float* __restrict__ item_emb,
                                                 float* __restrict__ acc,
                                                 float* __restrict__ xcur) {
  size_t i = (size_t)blockIdx.x * TPB + threadIdx.x;        // float4 index
  const size_t total = (size_t)N_NODES * EMB / 4;            // 3.2M
  if (i >= total) return;
  const size_t user_q = (size_t)NUM_USERS * EMB / 4;         // 1.6M
  float4 v = (i < user_q) ? ((const float4*)user_emb)[i]
                          : ((const float4*)item_emb)[i - user_q];
  ((float4*)acc)[i]  = v;
  ((float4*)xcur)[i] = v;
}

// ---------------- SpMM: y[row] += val * x[col], atomic scatter ----------------
// Block stages CHUNK edges of (rows, cols, vals) into LDS via async-LDS B128
// copies, double-buffered across grid-stride chunks. 16 lanes per edge, each
// lane handles 4 dims via a float4 gather (coalesced 256B per edge).
__global__ void __launch_bounds__(TPB) lgcn_spmm(const int* __restrict__ rows,
                                                 const int* __restrict__ cols,
                                                 const float* __restrict__ vals,
                                                 const float* __restrict__ x,
                                                 float* __restrict__ y,
                                                 int n_edges) {
  __shared__ int   s_rows[2][CHUNK];
  __shared__ int   s_cols[2][CHUNK];
  __shared__ float s_vals[2][CHUNK];

  const int t       = threadIdx.x;
  const int nchunks = (n_edges + CHUNK - 1) / CHUNK;
  const int stride  = gridDim.x;

  auto chunk_full = [&](int chunk) -> bool {
    return (chunk + 1) * CHUNK <= n_edges;
  };

  auto stage = [&](int chunk, int b) {
    const int base = chunk * CHUNK + t * 4;
    if (chunk_full(chunk)) {                 // uniform per chunk: 3 async ops/wave
      async_cp16(&s_rows[b][t * 4], rows + base);
      async_cp16(&s_cols[b][t * 4], cols + base);
      async_cp16(&s_vals[b][t * 4], vals + base);
    } else {                                 // tail chunk: plain loads, val=0 pad
      for (int j = 0; j < 4; ++j) {
        int idx = base + j;
        bool ok = idx < n_edges;
        s_rows[b][t * 4 + j] = ok ? rows[idx] : 0;
        s_cols[b][t * 4 + j] = ok ? cols[idx] : 0;
        s_vals[b][t * 4 + j] = ok ? vals[idx] : 0.0f;
      }
    }
  };

  int chunk = blockIdx.x;
  if (chunk < nchunks) stage(chunk, 0);

  const int q4    = (t & 15) * 4;   // dim quad -> dims [4q, 4q+3]
  const int eslot = t >> 4;         // 16 edges in flight per inner iteration
  int buf = 0;

  for (; chunk < nchunks; chunk += stride, buf ^= 1) {
    const int nxt = chunk + stride;
    const bool nxt_async = (nxt < nchunks) && chunk_full(nxt);
    if (nxt < nchunks) stage(nxt, buf ^ 1);
    if (nxt_async) {
      async_wait<3>();   // current buf's (older) 3 loads retired, 3 newer in flight
    } else {
      async_wait<0>();   // drain everything (tail / last chunk)
    }
    __syncthreads();

#pragma unroll 4
    for (int it = 0; it < CHUNK / 16; ++it) {
      const int   el = it * 16 + eslot;
      const int   r  = s_rows[buf][el];
      const int   c  = s_cols[buf][el];
      const float v  = s_vals[buf][el];
      const float4 xv = *(const float4*)(x + (size_t)c * EMB + q4);
      float* yp = y + (size_t)r * EMB + q4;
      atom_add_f32(yp + 0, v * xv.x);
      atom_add_f32(yp + 1, v * xv.y);
      atom_add_f32(yp + 2, v * xv.z);
      atom_add_f32(yp + 3, v * xv.w);
    }
    __syncthreads();   // buf may be re-staged next iteration
  }
}

// ---------------- accumulate via WMMA: acc = scale * (acc + x) ----------------
// One wave per 16(node) x 16(dim) tile.  D = (scale*I16) x Xtile + (scale*acc),
// contracted over K=16 with 4 chained v_wmma_f32_16x16x4_f32.  A and B are
// built with the same assumed K layout, so any hardware K-permutation cancels.
__global__ void __launch_bounds__(TPB) lgcn_accum(float* __restrict__ acc,
                                                  const float* __restrict__ x,
                                                  float scale) {
  const int wave  = (int)((blockIdx.x * (unsigned)TPB + threadIdx.x) >> 5);
  const int lane  = threadIdx.x & 31;
  const int tiles_d = EMB / 16;                 // 4
  const int node0 = (wave / tiles_d) * 16;
  const int d0    = (wave % tiles_d) * 16;
  if (node0 >= N_NODES) return;                 // wave-uniform: EXEC stays all-1s

  const int lhi = lane >> 4;                    // lane group 0/1
  const int llo = lane & 15;

#if __has_builtin(__builtin_amdgcn_wmma_f32_16x16x4_f32)
  // C = scale * acc tile, in C/D layout: VGPR r <-> row M = r + 8*lhi, col N = llo
  v8f c;
#pragma unroll
  for (int r = 0; r < 8; ++r)
    c[r] = scale * acc[(size_t)(node0 + r + 8 * lhi) * EMB + d0 + llo];

  v8f d = c;
#pragma unroll
  for (int kc = 0; kc < 4; ++kc) {
    // B chunk (4x16): VGPR r <-> K = r + 2*lhi, N = llo ; rows 4kc..4kc+3 of X
    v2f b;
    b[0] = x[(size_t)(node0 + 4 * kc + 0 + 2 * lhi) * EMB + d0 + llo];
    b[1] = x[(size_t)(node0 + 4 * kc + 1 + 2 * lhi) * EMB + d0 + llo];
    // A chunk (16x4) of scale*I16: VGPR r <-> k = r + 2*lhi, M = llo
    v2f a;
    a[0] = (llo == 4 * kc + 0 + 2 * lhi) ? scale : 0.0f;
    a[1] = (llo == 4 * kc + 1 + 2 * lhi) ? scale : 0.0f;
    d = __builtin_amdgcn_wmma_f32_16x16x4_f32(false, a, false, b,
                                              (short)0, d, false, false);
  }

#pragma unroll
  for (int r = 0; r < 8; ++r)
    acc[(size_t)(node0 + r + 8 * lhi) * EMB + d0 + llo] = d[r];
#else
  // Scalar fallback (same memory pattern)
#pragma unroll
  for (int r = 0; r < 8; ++r) {
    size_t idx = (size_t)(node0 + r + 8 * lhi) * EMB + d0 + llo;
    acc[idx] = scale * (acc[idx] + x[idx]);
  }
#endif
}

// ---------------- launcher ----------------
extern "C" void kernel_launch(void* const* d_in, const int* in_sizes, int n_in,
                              void* d_out, int out_size, void* d_ws, size_t ws_size,
                              hipStream_t stream) {
  const float* user_emb = (const float*)d_in[0];
  const float* item_emb = (const float*)d_in[1];
  const int*   rows     = (const int*)d_in[2];
  const int*   cols     = (const int*)d_in[3];
  const float* vals     = (const float*)d_in[4];
  const int    n_edges  = in_sizes[2];

  float* out   = (float*)d_out;                       // acc, doubles as output
  float* x_cur = (float*)d_ws;
  float* x_nxt = x_cur + (size_t)N_NODES * EMB;

  // out = x_cur = concat(user_emb, item_emb)
  {
    size_t nq = (size_t)N_NODES * EMB / 4;
    lgcn_init<<<(int)((nq + TPB - 1) / TPB), TPB, 0, stream>>>(user_emb, item_emb,
                                                               out, x_cur);
  }

  const int nchunks = (n_edges + CHUNK - 1) / CHUNK;
  const int spmm_blocks = nchunks < 1280 ? nchunks : 1280;
  const int accum_blocks = (N_NODES / 16) * (EMB / 16) * 32 / TPB;  // 6250

  for (int layer = 0; layer < NUM_LAYERS; ++layer) {
    (void)hipMemsetAsync(x_nxt, 0, (size_t)N_NODES * EMB * sizeof(float), stream);
    lgcn_spmm<<<spmm_blocks, TPB, 0, stream>>>(rows, cols, vals, x_cur, x_nxt,
                                               n_edges);
    const float scale = (layer == NUM_LAYERS - 1) ? 0.25f : 1.0f;
    lgcn_accum<<<accum_blocks, TPB, 0, stream>>>(out, x_nxt, scale);
    float* tmp = x_cur; x_cur = x_nxt; x_nxt = tmp;
  }
}